// GeneExpressionGNN_82944408420780
// MI455X (gfx1250) — compile-verified
//
#include <hip/hip_runtime.h>
#include <hip/hip_bf16.h>

// ---------------------------------------------------------------------------
// GCN (2-layer) for MI455X / gfx1250.
// Memory-bound: edge gather/scatter dominates (~2.4 GB -> ~105us @ 23.3 TB/s).
// Working set (h/agg buffers, 150MB total live peak ~110MB) fits in 192MB L2,
// so scatter atomics resolve on-die. GEMMs use V_WMMA_F32_16X16X4_F32 (full
// f32 precision; compute is negligible so the f32 WMMA shape is ideal).
// ---------------------------------------------------------------------------

typedef __attribute__((ext_vector_type(2))) float v2f;
typedef __attribute__((ext_vector_type(8))) float v8f;

#define NUM_GRAPHS 64

// --- init: deg = 1.0 (self loop), zero pool + counts -----------------------
__global__ void gcn_init_kernel(float* deg, float* pool, float* cnt, int n) {
    int i = blockIdx.x * blockDim.x + threadIdx.x;
    if (i < n) deg[i] = 1.0f;
    if (i < NUM_GRAPHS * 64) pool[i] = 0.0f;
    if (i < NUM_GRAPHS) cnt[i] = 0.0f;
}

// --- deg[dst] += 1 over edges ---------------------------------------------
__global__ void gcn_deg_kernel(const int* __restrict__ dst, float* deg, int E) {
    int e = blockIdx.x * blockDim.x + threadIdx.x;
    if (e < E) atomicAdd(&deg[dst[e]], 1.0f);
}

// --- dinv = rsqrt(deg) (deg >= 1 always) ----------------------------------
__global__ void gcn_dinv_kernel(const float* __restrict__ deg, float* dinv, int n) {
    int i = blockIdx.x * blockDim.x + threadIdx.x;
    if (i < n) dinv[i] = __frsqrt_rn(deg[i]);
}

// --- norm[e] = dinv[src] * dinv[dst] --------------------------------------
__global__ void gcn_norm_kernel(const int* __restrict__ src, const int* __restrict__ dst,
                                const float* __restrict__ dinv, float* norm, int E) {
    int e = blockIdx.x * blockDim.x + threadIdx.x;
    if (e < E) norm[e] = dinv[src[e]] * dinv[dst[e]];
}

// --- C[M x N] = op(A[M x K]) @ W[K x N], one wave per 16x16 tile -----------
// op(A) = FUSE ? relu(A + bias) : A.  Uses V_WMMA_F32_16X16X4_F32.
// f32 A-frag layout: lanes 0-15 -> M=lane, {K=k0,k0+1}; lanes 16-31 -> {k0+2,k0+3}
// f32 B-frag layout: lanes 0-15 -> N=lane, {K=k0,k0+1}; lanes 16-31 -> {k0+2,k0+3}
// C layout: VGPR r, lane l -> row = r + (l>=16 ? 8 : 0), col = l&15
template <bool FUSE>
__global__ void gcn_gemm_wmma_kernel(const float* __restrict__ A,
                                     const float* __restrict__ W,
                                     const float* __restrict__ bias,
                                     float* __restrict__ C,
                                     int tilesM, int tilesN, int N, int K) {
    int wave = (blockIdx.x * blockDim.x + threadIdx.x) >> 5;
    int lane = threadIdx.x & 31;
    if (wave >= tilesM * tilesN) return;  // uniform per wave: EXEC stays all-1s
    int tm = wave / tilesN;
    int tn = wave - tm * tilesN;

    int arow  = tm * 16 + (lane & 15);   // A row handled by this lane
    int bcol  = tn * 16 + (lane & 15);   // B column handled by this lane
    int khalf = (lane >> 4) << 1;        // 0 for lanes 0-15, 2 for lanes 16-31

    v8f acc = {};
    for (int k0 = 0; k0 < K; k0 += 4) {
        int ka = k0 + khalf;
        float a0 = A[(long)arow * K + ka];
        float a1 = A[(long)arow * K + ka + 1];
        if (FUSE) {
            a0 = fmaxf(a0 + bias[ka], 0.0f);
            a1 = fmaxf(a1 + bias[ka + 1], 0.0f);
        }
        v2f a, b;
        a.x = a0;
        a.y = a1;
        b.x = W[ka * N + bcol];
        b.y = W[(ka + 1) * N + bcol];
        acc = __builtin_amdgcn_wmma_f32_16x16x4_f32(
            /*neg_a=*/false, a, /*neg_b=*/false, b,
            /*c_mod=*/(short)0, acc, /*reuse_a=*/false, /*reuse_b=*/false);
    }

    int orow = tm * 16 + ((lane >> 4) << 3);
    int ocol = tn * 16 + (lane & 15);
#pragma unroll
    for (int r = 0; r < 8; ++r)
        C[(long)(orow + r) * N + ocol] = acc[r];
}

// --- agg[i][d] = h[i][d] * dinv[i]^2 (self-loop term, also zero-init) ------
__global__ void gcn_selfloop_kernel(const float* __restrict__ h,
                                    const float* __restrict__ dinv,
                                    float* __restrict__ agg, int total, int dshift) {
    int t = blockIdx.x * blockDim.x + threadIdx.x;
    if (t >= total) return;
    int node = t >> dshift;
    float di = dinv[node];
    agg[t] = h[t] * di * di;
}

// --- edge scatter, D=128: one wave per edge, float4 per lane ---------------
__global__ void gcn_scatter128_kernel(const int* __restrict__ src, const int* __restrict__ dst,
                                      const float* __restrict__ norm,
                                      const float* __restrict__ h, float* __restrict__ agg,
                                      int E) {
    int t = blockIdx.x * blockDim.x + threadIdx.x;
    int e = t >> 5;
    int lane = t & 31;
    if (e >= E) return;
    float w = norm[e];
    int s = src[e], d = dst[e];
    float4 v = *(const float4*)(h + (long)s * 128 + lane * 4);
    float* ap = agg + (long)d * 128 + lane * 4;
    atomicAdd(ap + 0, w * v.x);
    atomicAdd(ap + 1, w * v.y);
    atomicAdd(ap + 2, w * v.z);
    atomicAdd(ap + 3, w * v.w);
}

// --- edge scatter, D=64: one wave per edge, float2 per lane ----------------
__global__ void gcn_scatter64_kernel(const int* __restrict__ src, const int* __restrict__ dst,
                                     const float* __restrict__ norm,
                                     const float* __restrict__ h, float* __restrict__ agg,
                                     int E) {
    int t = blockIdx.x * blockDim.x + threadIdx.x;
    int e = t >> 5;
    int lane = t & 31;
    if (e >= E) return;
    float w = norm[e];
    int s = src[e], d = dst[e];
    float2 v = *(const float2*)(h + (long)s * 64 + lane * 2);
    float* ap = agg + (long)d * 64 + lane * 2;
    atomicAdd(ap + 0, w * v.x);
    atomicAdd(ap + 1, w * v.y);
}

// --- mean-pool accumulate: relu(agg2 + b2) into pool[graph][dim] -----------
__global__ void gcn_pool_kernel(const int* __restrict__ batch,
                                const float* __restrict__ agg2,
                                const float* __restrict__ b2,
                                float* pool, float* cnt, int n) {
    int t = blockIdx.x * blockDim.x + threadIdx.x;
    if (t >= n * 64) return;
    int node = t >> 6;
    int d = t & 63;
    int g = batch[node];
    float v = fmaxf(agg2[t] + b2[d], 0.0f);
    atomicAdd(&pool[g * 64 + d], v);
    if (d == 0) atomicAdd(&cnt[g], 1.0f);
}

// --- out = pool / max(cnt, 1) ----------------------------------------------
__global__ void gcn_final_kernel(const float* __restrict__ pool,
                                 const float* __restrict__ cnt,
                                 float* __restrict__ out) {
    int t = blockIdx.x * blockDim.x + threadIdx.x;
    if (t >= NUM_GRAPHS * 64) return;
    out[t] = pool[t] / fmaxf(cnt[t >> 6], 1.0f);
}

static inline int ceil_div(long a, int b) { return (int)((a + b - 1) / b); }

extern "C" void kernel_launch(void* const* d_in, const int* in_sizes, int n_in,
                              void* d_out, int out_size, void* d_ws, size_t ws_size,
                              hipStream_t stream) {
    const float* x  = (const float*)d_in[0];
    const int* ei   = (const int*)d_in[1];
    const int* batch = (const int*)d_in[2];
    const float* W1 = (const float*)d_in[3];
    const float* b1 = (const float*)d_in[4];
    const float* W2 = (const float*)d_in[5];
    const float* b2 = (const float*)d_in[6];
    float* out = (float*)d_out;

    const int N = in_sizes[2];          // 100000 nodes
    const int E = in_sizes[1] / 2;      // 1.6M edges
    const int* src = ei;                // edge_index[0]
    const int* dst = ei + E;            // edge_index[1]

    // workspace layout (floats)
    float* ws = (float*)d_ws;
    size_t off = 0;
    float* deg  = ws + off; off += N;
    float* dinv = ws + off; off += N;
    float* norm = ws + off; off += E;
    float* pool = ws + off; off += NUM_GRAPHS * 64;
    float* cnt  = ws + off; off += NUM_GRAPHS;
    off = (off + 3) & ~(size_t)3;                  // float4 alignment
    float* bufA = ws + off; off += (size_t)N * 128; // h1; later h2 + agg2
    float* bufB = ws + off; off += (size_t)N * 128; // agg1
    float* h1   = bufA;
    float* agg1 = bufB;
    float* h2   = bufA;                   // reuses dead h1 (first N*64)
    float* agg2 = bufA + (size_t)N * 64;  // second half of bufA

    const int B = 256;

    // 1) degrees / normalization
    gcn_init_kernel<<<ceil_div(N, B), B, 0, stream>>>(deg, pool, cnt, N);
    gcn_deg_kernel<<<ceil_div(E, B), B, 0, stream>>>(dst, deg, E);
    gcn_dinv_kernel<<<ceil_div(N, B), B, 0, stream>>>(deg, dinv, N);
    gcn_norm_kernel<<<ceil_div(E, B), B, 0, stream>>>(src, dst, dinv, norm, E);

    // 2) layer 1: h1 = x @ W1   (M=100000 -> 6250 tiles, N=128 -> 8 tiles)
    {
        int tilesM = N / 16, tilesN = 128 / 16;
        long waves = (long)tilesM * tilesN;
        gcn_gemm_wmma_kernel<false><<<ceil_div(waves * 32, B), B, 0, stream>>>(
            x, W1, nullptr, h1, tilesM, tilesN, 128, 128);
    }
    gcn_selfloop_kernel<<<ceil_div((long)N * 128, B), B, 0, stream>>>(
        h1, dinv, agg1, N * 128, 7);
    gcn_scatter128_kernel<<<ceil_div((long)E * 32, B), B, 0, stream>>>(
        src, dst, norm, h1, agg1, E);

    // 3) layer 2: h2 = relu(agg1 + b1) @ W2  (bias+relu fused into A-load)
    {
        int tilesM = N / 16, tilesN = 64 / 16;
        long waves = (long)tilesM * tilesN;
        gcn_gemm_wmma_kernel<true><<<ceil_div(waves * 32, B), B, 0, stream>>>(
            agg1, W2, b1, h2, tilesM, tilesN, 64, 128);
    }
    gcn_selfloop_kernel<<<ceil_div((long)N * 64, B), B, 0, stream>>>(
        h2, dinv, agg2, N * 64, 6);
    gcn_scatter64_kernel<<<ceil_div((long)E * 32, B), B, 0, stream>>>(
        src, dst, norm, h2, agg2, E);

    // 4) global mean pool (relu(agg2 + b2))
    gcn_pool_kernel<<<ceil_div((long)N * 64, B), B, 0, stream>>>(
        batch, agg2, b2, pool, cnt, N);
    gcn_final_kernel<<<ceil_div(NUM_GRAPHS * 64, B), B, 0, stream>>>(pool, cnt, out);
}